// SHGCN_90340342104105
// MI455X (gfx1250) — compile-verified
//
#include <hip/hip_runtime.h>
#include <math.h>

#define DIM 128

typedef float v2f __attribute__((ext_vector_type(2)));
typedef float v8f __attribute__((ext_vector_type(8)));

// ---------------------------------------------------------------------------
// Kernel 1: zero the padded aggregation buffer (d_ws is poisoned by harness)
// ---------------------------------------------------------------------------
__global__ void SHGCN_zero_kernel(float4* __restrict__ p, int n4) {
    int i = blockIdx.x * blockDim.x + threadIdx.x;
    if (i < n4) p[i] = make_float4(0.f, 0.f, 0.f, 0.f);
}

// ---------------------------------------------------------------------------
// Kernel 2: edge-parallel SpMM scatter.
// One wave32 per edge: each lane gathers float4 (4*32 = 128 = DIM) of
// input[col], scales by val, and scatters with hardware fp32 atomics into
// agg[row]. input/agg are L2-resident (51.2 MB each vs 192 MB L2).
// ---------------------------------------------------------------------------
__global__ void SHGCN_spmm_kernel(const float* __restrict__ x,
                                  const int*   __restrict__ rows,
                                  const int*   __restrict__ cols,
                                  const float* __restrict__ vals,
                                  float*       __restrict__ agg,
                                  int nedges) {
    const int e = blockIdx.x * (blockDim.x >> 5) + (threadIdx.x >> 5);
    if (e >= nedges) return;
    const int lane = threadIdx.x & 31;

    const int   r = rows[e];
    const int   c = cols[e];
    const float v = vals[e];

    const float4 xi = *(const float4*)(x + (size_t)c * DIM + lane * 4);
    float* dst = agg + (size_t)r * DIM + lane * 4;

    unsafeAtomicAdd(dst + 0, v * xi.x);   // global_atomic_add_f32
    unsafeAtomicAdd(dst + 1, v * xi.y);
    unsafeAtomicAdd(dst + 2, v * xi.z);
    unsafeAtomicAdd(dst + 3, v * xi.w);
}

// ---------------------------------------------------------------------------
// Kernel 3: out = tanh(agg @ W^T + b) via V_WMMA_F32_16X16X4_F32.
// Block = 256 threads = 8 waves; wave w owns output N-tile [16w, 16w+16).
// Each wave preloads all 32 K-fragments of its B tile (W^T) once, then
// grid-strides over 16-row M-tiles.
//
// Fragment layouts (ISA 7.12.2, wave32, 32-bit data):
//   A 16x4:  lane<16  -> {A[m0+l, k0],   A[m0+l, k0+1]}
//            lane>=16 -> {A[m0+l, k0+2], A[m0+l, k0+3]}   (l = lane&15)
//   B 4x16:  B[k,n] = W[n0+n, k0+k]; mirrored half-wave split -> each lane
//            reads a contiguous float2 from row (n0+l) of W.
//   C/D 16x16 f32: vgpr v, lane l<16  -> (M=m0+v,   N=n0+l)
//                          lane l>=16 -> (M=m0+v+8, N=n0+l)
// ---------------------------------------------------------------------------
__global__ void SHGCN_gemm_tanh_kernel(const float* __restrict__ agg,
                                       const float* __restrict__ W,
                                       const float* __restrict__ bias,
                                       float*       __restrict__ out,
                                       int nnodes, int mtiles) {
    const int lane = threadIdx.x & 31;
    const int wave = threadIdx.x >> 5;        // 0..7 -> N-tile
    const int n0   = wave * 16;
    const int l    = lane & 15;
    const bool hi  = lane >= 16;
    const int kofs = hi ? 2 : 0;

    // Preload B fragments for this N-tile: B[k,n] = W[n0+n, k]
    v2f Bf[32];
#pragma unroll
    for (int k = 0; k < 32; ++k)
        Bf[k] = *(const v2f*)(W + (size_t)(n0 + l) * DIM + k * 4 + kofs);

    const float bv = bias[n0 + l];            // bias depends only on N

    for (int mt = blockIdx.x; mt < mtiles; mt += gridDim.x) {
        const int m0 = mt * 16;

        v8f c;
#pragma unroll
        for (int i = 0; i < 8; ++i) c[i] = bv;

        const float* arow = agg + (size_t)(m0 + l) * DIM + kofs;
#pragma unroll
        for (int k = 0; k < 32; ++k) {
            v2f a = *(const v2f*)(arow + k * 4);
            // 8 args: (neg_a, A, neg_b, B, c_mod, C, reuse_a, reuse_b)
            c = __builtin_amdgcn_wmma_f32_16x16x4_f32(
                    false, a, false, Bf[k], (short)0, c, false, false);
        }

#pragma unroll
        for (int v = 0; v < 8; ++v) {
            const int m = m0 + v + (hi ? 8 : 0);
            if (m < nnodes)
                out[(size_t)m * DIM + n0 + l] = tanhf(c[v]);
        }
    }
}

// ---------------------------------------------------------------------------
// Launch: inputs in setup_inputs() order:
//   0 input[N,128] f32, 1 adj_rows[E] int, 2 adj_cols[E] int,
//   3 adj_vals[E] f32, 4 W[128,128] f32, 5 b[128] f32
// d_ws holds agg padded to a multiple of 16 rows.
// ---------------------------------------------------------------------------
extern "C" void kernel_launch(void* const* d_in, const int* in_sizes, int n_in,
                              void* d_out, int out_size, void* d_ws, size_t ws_size,
                              hipStream_t stream) {
    const float* input = (const float*)d_in[0];
    const int*   rows  = (const int*)  d_in[1];
    const int*   cols  = (const int*)  d_in[2];
    const float* vals  = (const float*)d_in[3];
    const float* W     = (const float*)d_in[4];
    const float* bias  = (const float*)d_in[5];
    float* out = (float*)d_out;
    float* agg = (float*)d_ws;

    const int nnodes = in_sizes[0] / DIM;
    const int nedges = in_sizes[1];
    const int mtiles = (nnodes + 15) / 16;
    const int padElems = mtiles * 16 * DIM;

    {   // zero agg (padded)
        const int n4 = padElems / 4;
        SHGCN_zero_kernel<<<(n4 + 255) / 256, 256, 0, stream>>>((float4*)agg, n4);
    }
    {   // SpMM scatter: 8 waves (= 8 edges) per 256-thread block
        SHGCN_spmm_kernel<<<(nedges + 7) / 8, 256, 0, stream>>>(
            input, rows, cols, vals, agg, nedges);
    }
    {   // WMMA GEMM + bias + tanh
        const int blocks = mtiles < 1024 ? mtiles : 1024;
        SHGCN_gemm_tanh_kernel<<<blocks, 256, 0, stream>>>(
            agg, W, bias, out, nnodes, mtiles);
    }
}